// Model_65755949302071
// MI455X (gfx1250) — compile-verified
//
#include <hip/hip_runtime.h>

#define B_      32
#define C_IN    10
#define HW      65536      // 256*256
#define C_OUT   128
#define E_      8
#define TPW     16         // 16x16 pixel tiles per wave

typedef __attribute__((ext_vector_type(2))) float v2f;
typedef __attribute__((ext_vector_type(8))) float v8f;

// ---------------- Phase 1: global average pool ----------------
// one block per (b, c): 320 blocks x 256 threads, float4 streaming reads
__global__ void pool_kernel(const float* __restrict__ x, float* __restrict__ pooled) {
    __shared__ float red[256];
    const int bc = blockIdx.x;                       // b*C_IN + c
    const float4* p = (const float4*)(x + (size_t)bc * HW);
    float s = 0.f;
    for (int i = threadIdx.x; i < HW / 4; i += 256) {
        float4 v = p[i];
        s += v.x + v.y + v.z + v.w;
    }
    red[threadIdx.x] = s;
    __syncthreads();
    for (int off = 128; off > 0; off >>= 1) {
        if ((int)threadIdx.x < off) red[threadIdx.x] += red[threadIdx.x + off];
        __syncthreads();
    }
    if (threadIdx.x == 0) pooled[bc] = red[0] * (1.0f / (float)HW);
}

// ---------------- Phase 2: routing + per-sample kernel mix ----------------
// one block per batch, 128 threads (one per output channel)
__global__ void mix_kernel(const float* __restrict__ pooled,
                           const float* __restrict__ weight,   // [E, C_OUT, C_IN]
                           const float* __restrict__ route_w,  // [E, C_IN]
                           const float* __restrict__ route_b,  // [E]
                           float* __restrict__ Kmat) {          // [B, C_OUT, C_IN]
    const int b = blockIdx.x;
    __shared__ float route[E_];
    const int tid = threadIdx.x;
    if (tid < E_) {
        float s = route_b[tid];
        #pragma unroll
        for (int i = 0; i < C_IN; ++i) s += pooled[b * C_IN + i] * route_w[tid * C_IN + i];
        route[tid] = 1.0f / (1.0f + __expf(-s));   // sigmoid
    }
    __syncthreads();
    const int o = tid;   // 0..127
    #pragma unroll
    for (int i = 0; i < C_IN; ++i) {
        float s = 0.f;
        #pragma unroll
        for (int e = 0; e < E_; ++e) s += route[e] * weight[(e * C_OUT + o) * C_IN + i];
        Kmat[((size_t)b * C_OUT + o) * C_IN + i] = s;
    }
}

// ---------------- Phase 3: per-batch [128x12] @ [12x65536] via FP32 WMMA ----------------
// K padded 10 -> 12. Column 10 is zero; column 11 carries bias with a matching
// all-ones B row, so out = K_hat @ x_hat includes the bias and C starts at 0.
// Each wave: one batch, TPW consecutive 16-pixel tiles, ALL 128 output channels
// (8 M-tiles). x is read exactly once device-wide; out written once with NT hint.
__global__ void __launch_bounds__(256, 2)
condconv_wmma(const float* __restrict__ x,     // [B, C_IN, HW]
              const float* __restrict__ Kmat,  // [B, C_OUT, C_IN]
              const float* __restrict__ bias,  // [C_OUT]
              float* __restrict__ out) {        // [B, C_OUT, HW]
    const int lane = threadIdx.x & 31;
    const int wid  = (blockIdx.x * blockDim.x + threadIdx.x) >> 5;
    const int waves_per_batch = HW / (16 * TPW);   // 256
    const int b     = wid / waves_per_batch;
    const int strip = wid % waves_per_batch;
    const int hf    = lane >> 4;                   // half-wave select (0/1)
    const int l15   = lane & 15;

    // ---- A fragments (loop-invariant). A[16x4] f32 layout:
    //   lanes 0-15: M=lane, v.x=K0, v.y=K1 ; lanes 16-31: M=lane-16, v.x=K2, v.y=K3
    const float* Kb = Kmat + (size_t)b * C_OUT * C_IN;
    v2f afrag[8][3];
    #pragma unroll
    for (int mt = 0; mt < 8; ++mt) {
        const int m = mt * 16 + l15;
        #pragma unroll
        for (int kc = 0; kc < 2; ++kc) {           // k-chunks 0..3 and 4..7: plain loads
            const int k0 = kc * 4 + hf * 2;
            v2f a;
            a.x = Kb[m * C_IN + k0];
            a.y = Kb[m * C_IN + k0 + 1];
            afrag[mt][kc] = a;
        }
        // k-chunk 8..11: hf=0 -> (K8, K9); hf=1 -> (0, bias[m])   [branch-free]
        {
            const float  t0 = Kb[m * C_IN + 8];                    // always in bounds
            const float* py = hf ? (bias + m) : (Kb + m * C_IN + 9); // address select
            const float  ty = *py;
            v2f a;
            a.x = hf ? 0.0f : t0;
            a.y = ty;
            afrag[mt][2] = a;
        }
    }

    const float* xb = x   + (size_t)b * C_IN  * HW;
    float*       ob = out + (size_t)b * C_OUT * HW;
    const int n_base = strip * 16 * TPW;

    for (int t = 0; t < TPW; ++t) {
        const int n = n_base + t * 16 + l15;   // pixel index for this lane

        // B[4x16] f32 layout mirrors A: lane&15 = N, K = hf*2 + {v.x, v.y}
        v2f bfrag[3];
        #pragma unroll
        for (int kc = 0; kc < 2; ++kc) {
            const int k0 = kc * 4 + hf * 2;
            v2f bv;
            bv.x = xb[(size_t)k0       * HW + n];
            bv.y = xb[(size_t)(k0 + 1) * HW + n];
            bfrag[kc] = bv;
        }
        {   // k rows 8..11: hf=0 -> (x8, x9); hf=1 -> (0, 1)  [ones row pairs with bias col]
            const float t0 = xb[(size_t)8 * HW + n];
            const float t1 = xb[(size_t)9 * HW + n];
            v2f bv;
            bv.x = hf ? 0.0f : t0;
            bv.y = hf ? 1.0f : t1;
            bfrag[2] = bv;
        }

        // kc=0 starts from C=0 (bias arrives via the augmented k=11 column)
        v8f acc[8];
        #pragma unroll
        for (int mt = 0; mt < 8; ++mt) {
            v8f z = {};
            acc[mt] = __builtin_amdgcn_wmma_f32_16x16x4_f32(
                false, afrag[mt][0], false, bfrag[0], (short)0, z, false, false);
        }
        #pragma unroll
        for (int kc = 1; kc < 3; ++kc)
            #pragma unroll
            for (int mt = 0; mt < 8; ++mt)
                acc[mt] = __builtin_amdgcn_wmma_f32_16x16x4_f32(
                    false, afrag[mt][kc], false, bfrag[kc], (short)0, acc[mt], false, false);

        // store 128ch x 16px tile block: NT stores (out never re-read; keep x in L2)
        #pragma unroll
        for (int mt = 0; mt < 8; ++mt) {
            #pragma unroll
            for (int r = 0; r < 8; ++r) {
                const int ch = mt * 16 + hf * 8 + r;
                __builtin_nontemporal_store(acc[mt][r], &ob[(size_t)ch * HW + n]);
            }
        }
    }
}

extern "C" void kernel_launch(void* const* d_in, const int* in_sizes, int n_in,
                              void* d_out, int out_size, void* d_ws, size_t ws_size,
                              hipStream_t stream) {
    const float* x       = (const float*)d_in[0];
    const float* weight  = (const float*)d_in[1];
    const float* bias    = (const float*)d_in[2];
    const float* route_w = (const float*)d_in[3];
    const float* route_b = (const float*)d_in[4];
    float* out = (float*)d_out;

    float* Kmat   = (float*)d_ws;                          // B*C_OUT*C_IN = 40960 floats
    float* pooled = Kmat + (size_t)B_ * C_OUT * C_IN;      // +320 floats

    // Phase 1: GAP (also warms x into the 192MB L2 for phase 3)
    pool_kernel<<<B_ * C_IN, 256, 0, stream>>>(x, pooled);
    // Phase 2: routing + expert mix
    mix_kernel<<<B_, C_OUT, 0, stream>>>(pooled, weight, route_w, route_b, Kmat);
    // Phase 3: WMMA conv. 32 batches * 256 strips = 8192 waves = 1024 blocks of 256
    const int waves_per_batch = HW / (16 * TPW);
    const int total_threads   = B_ * waves_per_batch * 32;
    condconv_wmma<<<total_threads / 256, 256, 0, stream>>>(x, Kmat, bias, out);
}